// ForceNormalizedResidualLoss_12146167513826
// MI455X (gfx1250) — compile-verified
//
#include <hip/hip_runtime.h>
#include <hip/hip_bf16.h>
#include <stdint.h>

#define TB 256

// ---------------- CDNA5 async-to-LDS path (guarded) ----------------
#if defined(__AMDGCN__) && __has_builtin(__builtin_amdgcn_global_load_async_to_lds_b32)
#define USE_ASYNC 1
#else
#define USE_ASYNC 0
#endif

#if USE_ASYNC && __has_builtin(__builtin_amdgcn_global_load_async_to_lds_b128)
#define ASYNC_CONN 1
#else
#define ASYNC_CONN 0
#endif

#if defined(__AMDGCN__) && __has_builtin(__builtin_amdgcn_s_wait_asynccnt)
#define WAIT_ASYNC(n) __builtin_amdgcn_s_wait_asynccnt(n)
#else
#define WAIT_ASYNC(n) do {} while (0)
#endif

#if USE_ASYNC
typedef __attribute__((address_space(1))) int gint_t;
typedef __attribute__((address_space(3))) int lint_t;
__device__ __forceinline__ void async_b32(const float* g, float* l) {
  __builtin_amdgcn_global_load_async_to_lds_b32(
      (gint_t*)(void*)const_cast<float*>(g),
      (lint_t*)(void*)l,
      /*offset=*/0, /*cpol=*/0);
}
#if ASYNC_CONN
typedef int v4i_t __attribute__((ext_vector_type(4)));
typedef __attribute__((address_space(1))) v4i_t gv4i_t;
typedef __attribute__((address_space(3))) v4i_t lv4i_t;
__device__ __forceinline__ void async_b128(const long long* g, long long* l) {
  __builtin_amdgcn_global_load_async_to_lds_b128(
      (gv4i_t*)(void*)const_cast<long long*>(g),
      (lv4i_t*)(void*)l,
      /*offset=*/0, /*cpol=*/0);
}
#endif
#endif

// Non-returning hardware f32 atomic (global_atomic_add_f32, STOREcnt path)
__device__ __forceinline__ void atomicAddF(float* p, float v) {
#if defined(__AMDGCN__)
  unsafeAtomicAdd(p, v);
#else
  atomicAdd(p, v);
#endif
}

// ---------------- per-element scatter ----------------
__device__ __forceinline__ void elem_body(
    long long nA, long long nB, float uc, float tc,
    const float* __restrict__ pred,
    float L, float E, float A, float I, float c, float s,
    float* __restrict__ Fint)
{
  const float* pa = pred + 3 * nA;
  const float* pb = pred + 3 * nB;
  float uA0 = pa[0] * uc, uA1 = pa[1] * uc, uA2 = pa[2] * tc;
  float uB0 = pb[0] * uc, uB1 = pb[1] * uc, uB2 = pb[2] * tc;

  float EA = E * A, EI = E * I;
  float ea_l  = EA / L;
  float ei_l  = EI / L;
  float L2    = L * L;
  float ei_l2 = EI / L2;
  float ei_l3 = EI / (L2 * L);

  float uAl =  c * uA0 + s * uA1;
  float wAl = -s * uA0 + c * uA1;
  float thA = -uA2;
  float uBl =  c * uB0 + s * uB1;
  float wBl = -s * uB0 + c * uB1;
  float thB = -uB2;

  float dw = wAl - wBl;
  float f0 = ea_l * (uAl - uBl);
  float f1 = 12.0f * ei_l3 * dw + 6.0f * ei_l2 * (thA + thB);
  float f2 =  6.0f * ei_l2 * dw + ei_l * (4.0f * thA + 2.0f * thB);
  float f5 =  6.0f * ei_l2 * dw + ei_l * (2.0f * thA + 4.0f * thB);
  // f3 = -f0, f4 = -f1  =>  f_gB(x,y) = -f_gA(x,y)

  float gA0 = c * f0 - s * f1;
  float gA1 = s * f0 + c * f1;

  atomicAddF(Fint + 3 * nA + 0,  gA0);
  atomicAddF(Fint + 3 * nA + 1,  gA1);
  atomicAddF(Fint + 3 * nA + 2,  -f2);
  atomicAddF(Fint + 3 * nB + 0, -gA0);
  atomicAddF(Fint + 3 * nB + 1, -gA1);
  atomicAddF(Fint + 3 * nB + 2,  -f5);
}

__global__ void elem_kernel(
    const float* __restrict__ pred, const float* __restrict__ ucp,
    const float* __restrict__ tcp,  const long long* __restrict__ conn,
    const float* __restrict__ len,  const float* __restrict__ pE,
    const float* __restrict__ pA,   const float* __restrict__ pI,
    const float* __restrict__ dir,  float* __restrict__ Fint, int nElem)
{
  const float uc = ucp[0], tc = tcp[0];
  const int tid = threadIdx.x;

#if USE_ASYNC
  // Double-buffered async staging of the per-element operand streams.
  __shared__ float sL[2][TB], sE[2][TB], sA[2][TB], sI[2][TB], sC[2][TB], sS[2][TB];
#if ASYNC_CONN
  __shared__ long long sConn[2][2 * TB];
  constexpr int NOPS = 7;
#else
  constexpr int NOPS = 6;
#endif
  const int numTiles = (nElem + TB - 1) / TB;

  auto issue = [&](int t, int b) {
    int e = t * TB + tid;
    if (e >= nElem) e = nElem - 1;        // clamp: keep EXEC uniform, loads harmless
    async_b32(len + e,         &sL[b][tid]);
    async_b32(pE + e,          &sE[b][tid]);
    async_b32(pA + e,          &sA[b][tid]);
    async_b32(pI + e,          &sI[b][tid]);
    async_b32(dir + 3 * e,     &sC[b][tid]);
    async_b32(dir + 3 * e + 2, &sS[b][tid]);
#if ASYNC_CONN
    async_b128(conn + 2 * e,   &sConn[b][2 * tid]);
#endif
  };

  int t = blockIdx.x;
  int buf = 0;
  if (t < numTiles) issue(t, 0);
  for (; t < numTiles; t += gridDim.x, buf ^= 1) {
    int nxt = t + gridDim.x;
    if (nxt < numTiles) {
      issue(nxt, buf ^ 1);
      WAIT_ASYNC(NOPS);                    // current tile's ops complete (in-order)
    } else {
      WAIT_ASYNC(0);
    }
    asm volatile("" ::: "memory");         // compiler barrier before LDS reads
    int e = t * TB + tid;
    if (e < nElem) {
#if ASYNC_CONN
      long long nA = sConn[buf][2 * tid + 0];
      long long nB = sConn[buf][2 * tid + 1];
#else
      long long nA = conn[2 * (long long)e + 0];
      long long nB = conn[2 * (long long)e + 1];
#endif
      elem_body(nA, nB, uc, tc, pred,
                sL[buf][tid], sE[buf][tid], sA[buf][tid], sI[buf][tid],
                sC[buf][tid], sS[buf][tid], Fint);
    }
  }
#else
  for (int e = blockIdx.x * TB + tid; e < nElem; e += gridDim.x * TB) {
    long long nA = conn[2 * (long long)e + 0];
    long long nB = conn[2 * (long long)e + 1];
    elem_body(nA, nB, uc, tc, pred,
              len[e], pE[e], pA[e], pI[e], dir[3 * e], dir[3 * e + 2], Fint);
  }
#endif
}

// ---------------- workspace zeroing ----------------
__global__ void zero_ws_kernel(float* __restrict__ f, int n, double* __restrict__ acc) {
  int i = blockIdx.x * blockDim.x + threadIdx.x;
  if (i < n) f[i] = 0.0f;
  if (i == 0) { acc[0] = 0.0; acc[1] = 0.0; }
}

// ---------------- per-node masked residual + reduction ----------------
__global__ void node_kernel(
    const float* __restrict__ Fint, const float* __restrict__ Fext,
    const int* __restrict__ bcd, const int* __restrict__ bcr,
    double* __restrict__ acc, int n)
{
  int i = blockIdx.x * blockDim.x + threadIdx.x;
  float sR = 0.0f, sF = 0.0f;
  if (i < n) {
    float fd = 1.0f - (float)bcd[i];
    float fr = 1.0f - (float)bcr[i];
    float fe0 = Fext[3 * i + 0], fe1 = Fext[3 * i + 1], fe2 = Fext[3 * i + 2];
    float r0 = (Fint[3 * i + 0] - fe0) * fd;
    float r1 = (Fint[3 * i + 1] - fe1) * fd;
    float r2 = (Fint[3 * i + 2] - fe2) * fr;
    float F0 = fe0 * fd, F1 = fe1 * fd, F2 = fe2 * fr;
    sR = r0 * r0 + r1 * r1 + r2 * r2;
    sF = F0 * F0 + F1 * F1 + F2 * F2;
  }
  // wave32 shuffle reduction
  for (int off = 16; off > 0; off >>= 1) {
    sR += __shfl_down(sR, off, 32);
    sF += __shfl_down(sF, off, 32);
  }
  __shared__ float wR[8], wF[8];
  int lane = threadIdx.x & 31, wave = threadIdx.x >> 5;
  if (lane == 0) { wR[wave] = sR; wF[wave] = sF; }
  __syncthreads();
  if (wave == 0) {
    sR = (lane < 8) ? wR[lane] : 0.0f;
    sF = (lane < 8) ? wF[lane] : 0.0f;
    for (int off = 4; off > 0; off >>= 1) {
      sR += __shfl_down(sR, off, 32);
      sF += __shfl_down(sF, off, 32);
    }
    if (lane == 0) {
      atomicAdd(&acc[0], (double)sR);
      atomicAdd(&acc[1], (double)sF);
    }
  }
}

__global__ void finalize_kernel(const double* __restrict__ acc, float* __restrict__ out) {
  double sF = acc[1];
  if (sF < 1e-30) sF = 1e-30;
  out[0] = (float)(acc[0] / sF);
}

// ---------------- launch ----------------
extern "C" void kernel_launch(void* const* d_in, const int* in_sizes, int n_in,
                              void* d_out, int out_size, void* d_ws, size_t ws_size,
                              hipStream_t stream)
{
  const float*     pred = (const float*)d_in[0];
  const float*     uc   = (const float*)d_in[1];
  const float*     tc   = (const float*)d_in[2];
  const long long* conn = (const long long*)d_in[3];
  const float*     len  = (const float*)d_in[4];
  const float*     pE   = (const float*)d_in[5];
  const float*     pA   = (const float*)d_in[6];
  const float*     pI   = (const float*)d_in[7];
  const float*     dir  = (const float*)d_in[8];
  const float*     Fext = (const float*)d_in[9];
  const int*       bcd  = (const int*)d_in[10];
  const int*       bcr  = (const int*)d_in[11];

  const int nNodes = in_sizes[0] / 3;
  const int nElem  = in_sizes[4];
  const int nF     = nNodes * 3;

  float* Fint = (float*)d_ws;
  size_t accOff = ((size_t)nF * sizeof(float) + 15) & ~(size_t)15;
  double* acc = (double*)((char*)d_ws + accOff);

  zero_ws_kernel<<<(nF + 255) / 256, 256, 0, stream>>>(Fint, nF, acc);

  int numTiles = (nElem + TB - 1) / TB;
  int G = numTiles < 2048 ? numTiles : 2048;   // >=2 tiles/block -> async double-buffer overlap
  elem_kernel<<<G, TB, 0, stream>>>(pred, uc, tc, conn, len, pE, pA, pI, dir, Fint, nElem);

  node_kernel<<<(nNodes + 255) / 256, 256, 0, stream>>>(Fint, Fext, bcd, bcr, acc, nNodes);

  finalize_kernel<<<1, 1, 0, stream>>>(acc, (float*)d_out);
}